// CrossAttention_12833362280629
// MI455X (gfx1250) — compile-verified
//
#include <hip/hip_runtime.h>
#include <math.h>

// Problem constants (match reference)
#define XD     256   // XDIM
#define NQ     512   // N queries
#define NM     512   // M keys/values
#define QB     8     // queries per thread in attention stage

typedef float v2f __attribute__((ext_vector_type(2)));
typedef float v8f __attribute__((ext_vector_type(8)));

// ---------------------------------------------------------------------------
// Stage 1: Y = (X @ W^T + bias) * scale      X:[R,256] W:[256,256] Y:[R,256]
// One wave per 16x16 output tile, V_WMMA_F32_16X16X4_F32, K stepped by 4.
//
// ISA layouts (wave32):
//   A 16x4 f32 : lanes 0-15 rows M=0..15, VGPR0=K0 VGPR1=K1;
//                lanes 16-31 same rows,   VGPR0=K2 VGPR1=K3.
//   B 4x16 f32 : lanes 0-15 cols N=0..15, VGPR0=K0 VGPR1=K1;
//                lanes 16-31 same cols,   VGPR0=K2 VGPR1=K3.
//   C/D 16x16  : VGPR v, lanes 0-15 -> row v, col=lane; lanes16-31 -> row v+8.
// => each lane loads a contiguous float2 from X and from W per K-step.
// ---------------------------------------------------------------------------
__global__ __launch_bounds__(256)
void qkv_gemm_wmma(const float* __restrict__ X, const float* __restrict__ W,
                   const float* __restrict__ bias, float* __restrict__ Y,
                   float scale) {
    const int lane   = threadIdx.x & 31;
    const int wave   = threadIdx.x >> 5;
    const int tile_m = blockIdx.x;                 // 16-row tile of X/Y
    const int tile_n = blockIdx.y * 8 + wave;      // 16-col tile of Y (= rows of W)
    const int hi     = lane >> 4;                  // half-wave: 0 or 1
    const int l16    = lane & 15;

    const int arow = tile_m * 16 + l16;            // X row this lane feeds
    const int brow = tile_n * 16 + l16;            // W row (output column)
    const int koff = hi * 2;                       // lanes16-31 handle K+2,K+3

    const float* __restrict__ Aptr = X + arow * XD + koff;
    const float* __restrict__ Bptr = W + brow * XD + koff;

    v8f acc = {};
#pragma unroll 8
    for (int k = 0; k < XD; k += 4) {
        v2f a = *(const v2f*)(Aptr + k);           // 8B aligned: koff,k even
        v2f b = *(const v2f*)(Bptr + k);
        acc = __builtin_amdgcn_wmma_f32_16x16x4_f32(
            /*neg_a=*/false, a, /*neg_b=*/false, b,
            /*c_mod=*/(short)0, acc, /*reuse_a=*/false, /*reuse_b=*/false);
    }

    const float bb = bias[tile_n * 16 + l16];
#pragma unroll
    for (int v = 0; v < 8; ++v) {
        const int row = tile_m * 16 + v + hi * 8;
        Y[row * XD + tile_n * 16 + l16] = (acc[v] + bb) * scale;
    }
}

// ---------------------------------------------------------------------------
// Stage 2: per-channel softmax attention.
//   out[n,j] = sum_m exp2(Qs[n,j]*K[m,j]) * V[m,j] / sum_m exp2(Qs[n,j]*K[m,j])
// Qs already carries log2(e)/sqrt(DF), so the native v_exp_f32 (2^x) applies.
// Shift-invariance of softmax + bounded inputs (|arg| << 128) => no max pass.
// Thread j handles channels j for QB queries; K/V loads are coalesced b32 and
// amortized over QB online accumulators.
// ---------------------------------------------------------------------------
__global__ __launch_bounds__(256)
void attn_perchannel(const float* __restrict__ Qs, const float* __restrict__ Kp,
                     const float* __restrict__ Vp, float* __restrict__ out) {
    const int j  = threadIdx.x;        // channel 0..255
    const int n0 = blockIdx.x * QB;    // first query of this block

    float q[QB], se[QB], ac[QB];
#pragma unroll
    for (int i = 0; i < QB; ++i) {
        q[i]  = Qs[(n0 + i) * XD + j];
        se[i] = 0.0f;
        ac[i] = 0.0f;
    }

    for (int m = 0; m < NM; ++m) {
        const float kv = Kp[m * XD + j];
        const float vv = Vp[m * XD + j];
#pragma unroll
        for (int i = 0; i < QB; ++i) {
            const float e = __builtin_amdgcn_exp2f(q[i] * kv);  // v_exp_f32
            se[i] += e;
            ac[i] += e * vv;
        }
    }

#pragma unroll
    for (int i = 0; i < QB; ++i)
        out[(n0 + i) * XD + j] = ac[i] * __builtin_amdgcn_rcpf(se[i]);
}

// ---------------------------------------------------------------------------
// Launch: 3 WMMA GEMMs (Q folds log2e/sqrt(32)), then attention.
// ---------------------------------------------------------------------------
extern "C" void kernel_launch(void* const* d_in, const int* in_sizes, int n_in,
                              void* d_out, int out_size, void* d_ws, size_t ws_size,
                              hipStream_t stream) {
    const float* x  = (const float*)d_in[0];   // [512,256]
    const float* c  = (const float*)d_in[1];   // [512,256]
    const float* Wq = (const float*)d_in[2];   // [256,256]
    const float* bq = (const float*)d_in[3];   // [256]
    const float* Wk = (const float*)d_in[4];
    const float* bk = (const float*)d_in[5];
    const float* Wv = (const float*)d_in[6];
    const float* bv = (const float*)d_in[7];
    float* out = (float*)d_out;                // [512,256]

    float* Q = (float*)d_ws;                   // 512*256 f32
    float* K = Q + NQ * XD;
    float* V = K + NM * XD;

    // exp(s/sqrt(32)) == exp2(s * log2(e)/sqrt(32)); fold into Q.
    const float qscale = 1.4426950408889634f / sqrtf(32.0f);

    dim3 gGemm(NQ / 16, XD / (16 * 8));        // (32, 2)
    dim3 bGemm(256);                           // 8 waves -> 8 col tiles
    qkv_gemm_wmma<<<gGemm, bGemm, 0, stream>>>(x, Wq, bq, Q, qscale);
    qkv_gemm_wmma<<<gGemm, bGemm, 0, stream>>>(c, Wk, bk, K, 1.0f);
    qkv_gemm_wmma<<<gGemm, bGemm, 0, stream>>>(c, Wv, bv, V, 1.0f);

    dim3 gAttn(NQ / QB);                       // 64 blocks
    dim3 bAttn(XD);                            // 256 threads: one per channel
    attn_perchannel<<<gAttn, bAttn, 0, stream>>>(Q, K, V, out);
}